// BaseNeuralNetworkPotential_54932631716420
// MI455X (gfx1250) — compile-verified
//
#include <hip/hip_runtime.h>
#include <hip/hip_bf16.h>

// ---- problem constants (match reference) ----
#define N_ATOMS 6000
#define NB      375                 // N_ATOMS / 16, exact
#define NPAIRS  17997000LL          // N*(N-1)/2
#define CUT2_F  0.25f               // CUTOFF^2

typedef float v2f __attribute__((ext_vector_type(2)));
typedef float v8f __attribute__((ext_vector_type(8)));

// Emit one pair (i<j guaranteed by caller). All index math is 32-bit
// (P < 2^31, 3P*4B < 2^31), stores are non-temporal (output is write-once,
// never re-read; keep L2 for the position working set).
__device__ __forceinline__ void emit_pair(
    int i, int j, float gval,
    float sqi, float xix, float xiy, float xiz,
    float sqj, float xjx, float xjy, float xjz,
    int mi, int mj,
    int* __restrict__ outI, int* __restrict__ outJ,
    float* __restrict__ outD, float* __restrict__ outR,
    float* __restrict__ outV)
{
    float d2 = fmaf(-2.0f, gval, sqi + sqj);   // |xi|^2 + |xj|^2 - 2 xi.xj
    d2 = fmaxf(d2, 0.0f);
    const bool valid = (mi == mj) && (d2 <= CUT2_F);   // cutoff on d^2
    const float d = __builtin_amdgcn_sqrtf(d2);        // raw v_sqrt_f32

    const float dd = valid ? d : 0.0f;
    const float rx = valid ? (xjx - xix) : 0.0f;
    const float ry = valid ? (xjy - xiy) : 0.0f;
    const float rz = valid ? (xjz - xiz) : 0.0f;

    // triu(k=1) linear pair index, 32-bit: p = i*(N-1) - i(i-1)/2 + j-i-1
    const unsigned p = (unsigned)(i * (N_ATOMS - 1) - ((i * (i - 1)) >> 1)
                                  + (j - i - 1));

    __builtin_nontemporal_store(i,  outI + p);
    __builtin_nontemporal_store(j,  outJ + p);
    __builtin_nontemporal_store(dd, outD + p);
    __builtin_nontemporal_store(rx, outR + 3u * p + 0u);
    __builtin_nontemporal_store(ry, outR + 3u * p + 1u);
    __builtin_nontemporal_store(rz, outR + 3u * p + 2u);
    __builtin_nontemporal_store(valid ? 1.0f : 0.0f, outV + p);
}

// One wave (32 lanes) per 16x16 (i,j) tile:
//   stage 16+16 atoms into LDS -> v_wmma_f32_16x16x4_f32 Gram tile ->
//   d^2 = |xi|^2+|xj|^2-2g epilogue -> coalesced NT stores.
__global__ __launch_bounds__(32)
void pairdist_wmma_kernel(const float* __restrict__ pos,
                          const int*   __restrict__ mol,
                          int*   __restrict__ outI,
                          int*   __restrict__ outJ,
                          float* __restrict__ outD,
                          float* __restrict__ outR,
                          float* __restrict__ outV)
{
    const int bj = blockIdx.x;
    const int bi = blockIdx.y;
    if (bi > bj) return;                 // upper-triangular tiles (uniform)

    const int i0 = bi * 16;
    const int j0 = bj * 16;

    __shared__ float posI[16][3];
    __shared__ float posJ[16][3];
    __shared__ float sqI[16];
    __shared__ float sqJ[16];
    __shared__ int   molI[16];
    __shared__ int   molJ[16];

    const int lane = threadIdx.x;        // 0..31, single wave32
    const int sub  = lane & 15;

    if (lane < 16) {
        const int ai = i0 + sub;
        const float x = pos[ai * 3 + 0], y = pos[ai * 3 + 1], z = pos[ai * 3 + 2];
        posI[sub][0] = x; posI[sub][1] = y; posI[sub][2] = z;
        sqI[sub]  = x * x + y * y + z * z;
        molI[sub] = mol[ai];
    } else {
        const int aj = j0 + sub;
        const float x = pos[aj * 3 + 0], y = pos[aj * 3 + 1], z = pos[aj * 3 + 2];
        posJ[sub][0] = x; posJ[sub][1] = y; posJ[sub][2] = z;
        sqJ[sub]  = x * x + y * y + z * z;
        molJ[sub] = mol[aj];
    }
    __syncthreads();

    // A-matrix 16x4 f32 layout (ISA 7.12.2): lanes 0-15 -> VGPR0=K0, VGPR1=K1;
    // lanes 16-31 -> VGPR0=K2, VGPR1=K3. B mirrors with N instead of M.
    const int khalf = (lane >> 4) << 1;          // 0 (x,y) or 2 (z,pad)
    v2f a, b;
    a.x = posI[sub][khalf];
    a.y = (khalf == 0) ? posI[sub][1] : 0.0f;    // pad K=3 with 0
    b.x = posJ[sub][khalf];
    b.y = (khalf == 0) ? posJ[sub][1] : 0.0f;

    v8f g = {};
    // 8 args: (neg_a, A, neg_b, B, c_mod, C, reuse_a, reuse_b)
    g = __builtin_amdgcn_wmma_f32_16x16x4_f32(false, a, false, b,
                                              (short)0, g, false, false);

    // C/D layout: VGPR v holds row m = v (lanes 0-15) or v+8 (lanes 16-31),
    // col n = lane & 15.
    const int   n   = sub;
    const int   j   = j0 + n;
    const float xjx = posJ[n][0], xjy = posJ[n][1], xjz = posJ[n][2];
    const float sqj = sqJ[n];
    const int   mj  = molJ[n];
    const int   mBase = (lane >> 4) << 3;        // 0 or 8

    if (bi != bj) {
        // Off-diagonal tile: i < j always -> no per-element exec masking.
#pragma unroll
        for (int v = 0; v < 8; ++v) {
            const int m = mBase + v;
            const int i = i0 + m;
            emit_pair(i, j, g[v],
                      sqI[m], posI[m][0], posI[m][1], posI[m][2],
                      sqj, xjx, xjy, xjz, molI[m], mj,
                      outI, outJ, outD, outR, outV);
        }
    } else {
        // Diagonal tile (375 of 70500): guard i < j.
#pragma unroll
        for (int v = 0; v < 8; ++v) {
            const int m = mBase + v;
            const int i = i0 + m;
            if (i < j) {
                emit_pair(i, j, g[v],
                          sqI[m], posI[m][0], posI[m][1], posI[m][2],
                          sqj, xjx, xjy, xjz, molI[m], mj,
                          outI, outJ, outD, outR, outV);
            }
        }
    }
}

extern "C" void kernel_launch(void* const* d_in, const int* in_sizes, int n_in,
                              void* d_out, int out_size, void* d_ws, size_t ws_size,
                              hipStream_t stream) {
    (void)in_sizes; (void)n_in; (void)out_size; (void)d_ws; (void)ws_size;

    const float* pos = (const float*)d_in[0];      // [6000,3] f32
    const int*   mol = (const int*)d_in[1];        // [6000] int32

    // Flat output: [I:P][J:P][d:P][r:3P][valid:P], 4-byte elements.
    int*   outI = (int*)d_out;
    int*   outJ = outI + NPAIRS;
    float* outD = (float*)d_out + 2 * NPAIRS;
    float* outR = (float*)d_out + 3 * NPAIRS;
    float* outV = (float*)d_out + 6 * NPAIRS;

    dim3 grid(NB, NB);                             // 375x375 tiles, bi<=bj kept
    pairdist_wmma_kernel<<<grid, 32, 0, stream>>>(pos, mol,
                                                  outI, outJ, outD, outR, outV);
}